// PowerFlowNet_1047972020764
// MI455X (gfx1250) — compile-verified
//
#include <hip/hip_runtime.h>

typedef float v2f __attribute__((ext_vector_type(2)));
typedef float v8f __attribute__((ext_vector_type(8)));

#define NNODES 20000
#define NEDGE  160000
#define H 128

// ---------------------------------------------------------------------------
// Generic 16x16-tiled GEMM:  Y[rows,M] = act( X[rows,K] @ W[K,M]
//                                             + bias[M] * (rowscale[row] or 1)
//                                             + addin[row,M] )
// One wave per 16x16 output tile using V_WMMA_F32_16X16X4_F32 (fp32 exact).
// rows multiple of 16, M multiple of 16, K multiple of 4.
// Epilogue configuration is compile-time to keep the epilogue branch-free.
// ---------------------------------------------------------------------------
template <int K, bool HAS_BIAS, bool ROWSCALE, bool HAS_ADD, bool RELU>
__global__ __launch_bounds__(256) void gemm16_kernel(
    const float* __restrict__ X, const float* __restrict__ W,
    const float* __restrict__ bias, const float* __restrict__ rowscale,
    const float* __restrict__ addin, float* __restrict__ Y,
    int rows, int M) {
  const int lane = threadIdx.x & 31;
  const int wid  = threadIdx.x >> 5;
  const int tilesM = M >> 4;
  const int numTiles = (rows >> 4) * tilesM;
  const int tile = blockIdx.x * 8 + wid;
  if (tile >= numTiles) return;  // wave-uniform; EXEC stays all-ones
  const int tm = (tile / tilesM) << 4;
  const int tn = (tile % tilesM) << 4;
  const int half = lane >> 4;   // 0: K pair {0,1}, 1: K pair {2,3} (A/B striping)
  const int l16  = lane & 15;

  const float* Xp = X + (tm + l16) * K;  // A: row m = tm + l16
  const float* Wp = W + tn + l16;        // B: col n = tn + l16

  v8f acc = {0.f, 0.f, 0.f, 0.f, 0.f, 0.f, 0.f, 0.f};
#pragma unroll
  for (int k0 = 0; k0 < K; k0 += 4) {
    const int ka = k0 + 2 * half;
    v2f a, b;
    a.x = Xp[ka];
    a.y = Xp[ka + 1];
    b.x = Wp[ka * M];
    b.y = Wp[(ka + 1) * M];
    acc = __builtin_amdgcn_wmma_f32_16x16x4_f32(false, a, false, b, (short)0,
                                                acc, false, false);
  }

  const int col = tn + l16;
  const float bc = HAS_BIAS ? bias[col] : 0.0f;
#pragma unroll
  for (int r = 0; r < 8; ++r) {
    const int row = tm + r + half * 8;  // C/D layout: VGPR r -> rows r, r+8
    float v = acc[r];
    if (HAS_BIAS) v += ROWSCALE ? rowscale[row] * bc : bc;
    if (HAS_ADD) v += addin[row * M + col];
    if (RELU) v = fmaxf(v, 0.0f);
    Y[row * M + col] = v;
  }
}

// ---------------------------------------------------------------------------
// Degree / normalization
// ---------------------------------------------------------------------------
__global__ __launch_bounds__(256) void deg_kernel(const int* __restrict__ src0,
                                                  const int* __restrict__ dst0,
                                                  float* __restrict__ deg) {
  int i = blockIdx.x * 256 + threadIdx.x;
  if (i < NEDGE) {
    atomicAdd(&deg[src0[i]], 1.0f);
    atomicAdd(&deg[dst0[i]], 1.0f);
  }
}

__global__ __launch_bounds__(256) void dis_kernel(const float* __restrict__ deg,
                                                  float* __restrict__ dis) {
  int i = blockIdx.x * 256 + threadIdx.x;
  if (i < NNODES) {
    float d = deg[i];
    dis[i] = d > 0.0f ? rsqrtf(d) : 0.0f;
  }
}

// ---------------------------------------------------------------------------
// EdgeAgg hidden scatter: one wave per directed edge (2E total).
// S[dst] += relu(nodeA[dst] + nodeB[src] + edgeC[j])     (b1 folded into edgeC)
// ---------------------------------------------------------------------------
__global__ __launch_bounds__(256) void ea_scatter_kernel(
    const float* __restrict__ nodeA, const float* __restrict__ nodeB,
    const float* __restrict__ edgeC, const int* __restrict__ src0,
    const int* __restrict__ dst0, float* __restrict__ S) {
  const int wv = (blockIdx.x * 256 + threadIdx.x) >> 5;
  const int lane = threadIdx.x & 31;
  if (wv >= 2 * NEDGE) return;
  const int j = (wv < NEDGE) ? wv : wv - NEDGE;
  int s = src0[j], d = dst0[j];
  if (wv >= NEDGE) { int t = s; s = d; d = t; }  // reversed copy
  float4 a = ((const float4*)(nodeA + d * H))[lane];
  float4 b = ((const float4*)(nodeB + s * H))[lane];
  float4 c = ((const float4*)(edgeC + j * H))[lane];
  float* out = S + d * H + lane * 4;
  atomicAdd(out + 0, fmaxf(a.x + b.x + c.x, 0.0f));
  atomicAdd(out + 1, fmaxf(a.y + b.y + c.y, 0.0f));
  atomicAdd(out + 2, fmaxf(a.z + b.z + c.z, 0.0f));
  atomicAdd(out + 3, fmaxf(a.w + b.w + c.w, 0.0f));
}

// ---------------------------------------------------------------------------
// TAGConv propagation hop: hnext[dst] += dis[src]*dis[dst] * hcur[src]
// ---------------------------------------------------------------------------
__global__ __launch_bounds__(256) void tag_hop_kernel(
    const float* __restrict__ hcur, const float* __restrict__ dis,
    const int* __restrict__ src0, const int* __restrict__ dst0,
    float* __restrict__ hnext) {
  const int wv = (blockIdx.x * 256 + threadIdx.x) >> 5;
  const int lane = threadIdx.x & 31;
  if (wv >= 2 * NEDGE) return;
  const int j = (wv < NEDGE) ? wv : wv - NEDGE;
  int s = src0[j], d = dst0[j];
  if (wv >= NEDGE) { int t = s; s = d; d = t; }
  const float nrm = dis[s] * dis[d];
  float4 v = ((const float4*)(hcur + s * H))[lane];
  float* out = hnext + d * H + lane * 4;
  atomicAdd(out + 0, nrm * v.x);
  atomicAdd(out + 1, nrm * v.y);
  atomicAdd(out + 2, nrm * v.z);
  atomicAdd(out + 3, nrm * v.w);
}

// ---------------------------------------------------------------------------
extern "C" void kernel_launch(void* const* d_in, const int* in_sizes, int n_in,
                              void* d_out, int out_size, void* d_ws,
                              size_t ws_size, hipStream_t stream) {
  (void)in_sizes; (void)n_in; (void)out_size; (void)ws_size;
  const float* x      = (const float*)d_in[0];
  const float* mask   = (const float*)d_in[1];
  const float* eattr  = (const float*)d_in[2];
  const int*   eidx   = (const int*)d_in[3];
  const float* m_w1   = (const float*)d_in[4];
  const float* m_b1   = (const float*)d_in[5];
  const float* m_w2   = (const float*)d_in[6];
  const float* m_b2   = (const float*)d_in[7];
  const float* ea0_w1 = (const float*)d_in[8];
  const float* ea0_b1 = (const float*)d_in[9];
  const float* ea0_w2 = (const float*)d_in[10];
  const float* ea0_b2 = (const float*)d_in[11];
  const float* tag0_w = (const float*)d_in[12];
  const float* tag0_b = (const float*)d_in[13];
  const float* ea1_w1 = (const float*)d_in[14];
  const float* ea1_b1 = (const float*)d_in[15];
  const float* ea1_w2 = (const float*)d_in[16];
  const float* ea1_b2 = (const float*)d_in[17];
  const float* tag1_w = (const float*)d_in[18];
  const float* tag1_b = (const float*)d_in[19];
  const float* ea2_w1 = (const float*)d_in[20];
  const float* ea2_b1 = (const float*)d_in[21];
  const float* ea2_w2 = (const float*)d_in[22];
  const float* ea2_b2 = (const float*)d_in[23];
  float* out = (float*)d_out;

  const int* src0 = eidx;
  const int* dst0 = eidx + NEDGE;

  // workspace layout (floats)
  float* ws  = (float*)d_ws;
  float* deg = ws;                  // N
  float* dis = deg + NNODES;        // N
  float* h16 = dis + NNODES;        // N*16
  float* nA  = h16 + NNODES * 16;   // N*128
  float* nB  = nA + NNODES * H;     // N*128
  float* S   = nB + NNODES * H;     // N*128
  float* P   = S + NNODES * H;      // N*128
  float* Q   = P + NNODES * H;      // N*128
  float* eC  = Q + NNODES * H;      // E*128

  auto gGrid = [](int rows, int M) {
    return dim3((((rows >> 4) * (M >> 4)) + 7) / 8);
  };
  const dim3 edgeGrid((2 * NEDGE * 32 + 255) / 256);
  const size_t nodeHbytes = (size_t)NNODES * H * sizeof(float);

  // ---- degrees + gcn norm ----
  hipMemsetAsync(deg, 0, NNODES * sizeof(float), stream);
  deg_kernel<<<(NEDGE + 255) / 256, 256, 0, stream>>>(src0, dst0, deg);
  dis_kernel<<<(NNODES + 255) / 256, 256, 0, stream>>>(deg, dis);

  // ---- mask embedding + residual:  h16 = relu(mask@w1+b1)@w2 + b2 + x ----
  gemm16_kernel<16, true, false, false, true>
      <<<gGrid(NNODES, H), 256, 0, stream>>>(mask, m_w1, m_b1, nullptr,
                                             nullptr, nA, NNODES, H);
  gemm16_kernel<H, true, false, true, false>
      <<<gGrid(NNODES, 16), 256, 0, stream>>>(nA, m_w2, m_b2, nullptr, x, h16,
                                              NNODES, 16);

  // EdgeAgg layer: hin[F] -> out[Mout]
  // w1 rows: [0,F) dst-part, [F,2F) src-part, [2F,2F+16) edge-part
  auto run_ea = [&](const float* hin, int F, const float* w1, const float* b1,
                    const float* w2, const float* b2, float* agg, float* outp,
                    int Mout, bool relu) {
    if (F == H) {
      gemm16_kernel<H, false, false, false, false>
          <<<gGrid(NNODES, H), 256, 0, stream>>>(hin, w1, nullptr, nullptr,
                                                 nullptr, nA, NNODES, H);
      gemm16_kernel<H, false, false, false, false>
          <<<gGrid(NNODES, H), 256, 0, stream>>>(hin, w1 + H * H, nullptr,
                                                 nullptr, nullptr, nB, NNODES,
                                                 H);
      gemm16_kernel<16, true, false, false, false>
          <<<gGrid(NEDGE, H), 256, 0, stream>>>(eattr, w1 + 2 * H * H, b1,
                                                nullptr, nullptr, eC, NEDGE, H);
    } else {  // F == 16
      gemm16_kernel<16, false, false, false, false>
          <<<gGrid(NNODES, H), 256, 0, stream>>>(hin, w1, nullptr, nullptr,
                                                 nullptr, nA, NNODES, H);
      gemm16_kernel<16, false, false, false, false>
          <<<gGrid(NNODES, H), 256, 0, stream>>>(hin, w1 + 16 * H, nullptr,
                                                 nullptr, nullptr, nB, NNODES,
                                                 H);
      gemm16_kernel<16, true, false, false, false>
          <<<gGrid(NEDGE, H), 256, 0, stream>>>(eattr, w1 + 32 * H, b1,
                                                nullptr, nullptr, eC, NEDGE, H);
    }
    hipMemsetAsync(agg, 0, nodeHbytes, stream);
    ea_scatter_kernel<<<edgeGrid, 256, 0, stream>>>(nA, nB, eC, src0, dst0,
                                                    agg);
    // out = act( agg @ w2 + deg*b2 )   (segment_sum pushed through linear w2)
    if (relu)
      gemm16_kernel<H, true, true, false, true>
          <<<gGrid(NNODES, Mout), 256, 0, stream>>>(agg, w2, b2, deg, nullptr,
                                                    outp, NNODES, Mout);
    else
      gemm16_kernel<H, true, true, false, false>
          <<<gGrid(NNODES, Mout), 256, 0, stream>>>(agg, w2, b2, deg, nullptr,
                                                    outp, NNODES, Mout);
  };

  // TAGConv: outp = relu( sum_k (L^k hin) @ lin_k + bias ), t1/t2 scratch
  auto run_tag = [&](const float* hin, const float* tw, const float* tb,
                     float* outp, float* t1, float* t2) {
    gemm16_kernel<H, true, false, false, false>
        <<<gGrid(NNODES, H), 256, 0, stream>>>(hin, tw, tb, nullptr, nullptr,
                                               outp, NNODES, H);
    hipMemsetAsync(t1, 0, nodeHbytes, stream);
    tag_hop_kernel<<<edgeGrid, 256, 0, stream>>>(hin, dis, src0, dst0, t1);
    gemm16_kernel<H, false, false, true, false>
        <<<gGrid(NNODES, H), 256, 0, stream>>>(t1, tw + H * H, nullptr,
                                               nullptr, outp, outp, NNODES, H);
    hipMemsetAsync(t2, 0, nodeHbytes, stream);
    tag_hop_kernel<<<edgeGrid, 256, 0, stream>>>(t1, dis, src0, dst0, t2);
    gemm16_kernel<H, false, false, true, false>
        <<<gGrid(NNODES, H), 256, 0, stream>>>(t2, tw + 2 * H * H, nullptr,
                                               nullptr, outp, outp, NNODES, H);
    hipMemsetAsync(t1, 0, nodeHbytes, stream);
    tag_hop_kernel<<<edgeGrid, 256, 0, stream>>>(t2, dis, src0, dst0, t1);
    gemm16_kernel<H, false, false, true, true>
        <<<gGrid(NNODES, H), 256, 0, stream>>>(t1, tw + 3 * H * H, nullptr,
                                               nullptr, outp, outp, NNODES, H);
  };

  // ---- pipeline ----
  run_ea(h16, 16, ea0_w1, ea0_b1, ea0_w2, ea0_b2, S, P, H, true);   // EA0 -> P
  run_tag(P, tag0_w, tag0_b, S, nA, nB);                            // TAG0 -> S
  run_ea(S, H, ea1_w1, ea1_b1, ea1_w2, ea1_b2, P, Q, H, true);      // EA1 -> Q
  run_tag(Q, tag1_w, tag1_b, P, nA, nB);                            // TAG1 -> P
  run_ea(P, H, ea2_w1, ea2_b1, ea2_w2, ea2_b2, S, out, 16, false);  // EA2->out
}